// CombinedLoss_50861002719695
// MI455X (gfx1250) — compile-verified
//
#include <hip/hip_runtime.h>
#include <hip/hip_bf16.h>
#include <stdint.h>

typedef __attribute__((ext_vector_type(2))) float v2f;
typedef __attribute__((ext_vector_type(8))) float v8f;

#define B_  4
#define N_  100000
#define T_  1000
#define TPAD_ 1024        // flow partials padded to a multiple of 64
#define HW_ 2048          // 32*64
#define PC_BLOCKS 512

// workspace layout (in floats)
#define WS_SCAL 0         // [0]=loss_transl [1]=loss_rot
#define WS_M    16        // 4 batches * 16 floats (9 rot + 3 trans)
#define WS_PC   128       // 512 pc partials
#define WS_FLOW 640       // 1024 flow partials (1000 real + 24 zeros)

// ---------------------------------------------------------------------------
// helpers
// ---------------------------------------------------------------------------
__device__ __forceinline__ float waveReduceSum(float v) {
    #pragma unroll
    for (int off = 16; off > 0; off >>= 1)
        v += __shfl_xor(v, off, 32);
    return v;
}

__device__ __forceinline__ void quat_to_rot(const float* q, float* R) {
    float w = q[0], x = q[1], y = q[2], z = q[3];
    float inv = rsqrtf(w*w + x*x + y*y + z*z);
    w *= inv; x *= inv; y *= inv; z *= inv;
    R[0] = 1.f - 2.f*(y*y + z*z); R[1] = 2.f*(x*y - z*w);       R[2] = 2.f*(x*z + y*w);
    R[3] = 2.f*(x*y + z*w);       R[4] = 1.f - 2.f*(x*x + z*z); R[5] = 2.f*(y*z - x*w);
    R[6] = 2.f*(x*z - y*w);       R[7] = 2.f*(y*z + x*w);       R[8] = 1.f - 2.f*(x*x + y*y);
}

// ---------------------------------------------------------------------------
// Kernel 1: scalar pose losses + per-batch M matrix (trivial work, B=4)
//   M = inv(RT_pred) @ RT_target = R_predT @ T(t_tgt - t_err) @ R_tgt
// ---------------------------------------------------------------------------
__global__ void setup_kernel(const float* __restrict__ tt, const float* __restrict__ tr,
                             const float* __restrict__ te, const float* __restrict__ re,
                             float* __restrict__ ws) {
    if (threadIdx.x != 0) return;
    // smooth-L1 translation loss
    float lt = 0.f;
    for (int b = 0; b < B_; ++b)
        for (int i = 0; i < 3; ++i) {
            float d = te[b*3+i] - tt[b*3+i];
            float a = fabsf(d);
            lt += (a < 1.f) ? 0.5f*d*d : (a - 0.5f);
        }
    lt *= (1.f / B_);
    // quaternion distance loss
    float lr = 0.f;
    for (int b = 0; b < B_; ++b) {
        float w1 = re[b*4+0], x1 = re[b*4+1], y1 = re[b*4+2], z1 = re[b*4+3];
        float w2 =  tr[b*4+0], x2 = -tr[b*4+1], y2 = -tr[b*4+2], z2 = -tr[b*4+3];
        float tw = w1*w2 - x1*x2 - y1*y2 - z1*z2;
        float tx = w1*x2 + x1*w2 + y1*z2 - z1*y2;
        float ty = w1*y2 - x1*z2 + y1*w2 + z1*x2;
        float tz = w1*z2 + x1*y2 - y1*x2 + z1*w2;
        float vn = sqrtf(tx*tx + ty*ty + tz*tz);
        lr += 2.f * atan2f(vn, fabsf(tw));
    }
    lr *= (1.f / B_);
    ws[WS_SCAL + 0] = lt;
    ws[WS_SCAL + 1] = lr;
    // M per batch
    for (int b = 0; b < B_; ++b) {
        float A[9], Bm[9];
        quat_to_rot(&re[b*4], A);   // R_pred
        quat_to_rot(&tr[b*4], Bm);  // R_target
        float d0 = tt[b*3+0] - te[b*3+0];
        float d1 = tt[b*3+1] - te[b*3+1];
        float d2 = tt[b*3+2] - te[b*3+2];
        float* M = &ws[WS_M + b*16];
        for (int i = 0; i < 3; ++i)
            for (int j = 0; j < 3; ++j)
                M[i*3+j] = A[0*3+i]*Bm[0*3+j] + A[1*3+i]*Bm[1*3+j] + A[2*3+i]*Bm[2*3+j];
        M[9]  = A[0]*d0 + A[3]*d1 + A[6]*d2;   // A^T d
        M[10] = A[1]*d0 + A[4]*d1 + A[7]*d2;
        M[11] = A[2]*d0 + A[5]*d1 + A[8]*d2;
    }
}

// ---------------------------------------------------------------------------
// Kernel 2: point-cloud loss partials. err = ||(M3-I)p + t||, summed.
// All batches share N so sum/N == Σ_b mean_b.
// ---------------------------------------------------------------------------
__global__ void pc_kernel(const float* __restrict__ pc, const float* __restrict__ ws,
                          float* __restrict__ partials) {
    __shared__ float sM[B_ * 12];
    __shared__ float swave[8];
    const int tid = threadIdx.x;
    if (tid < B_ * 12) sM[tid] = ws[WS_M + (tid / 12) * 16 + (tid % 12)];
    __syncthreads();

    float acc = 0.f;
    const int total = B_ * N_;
    for (int i = blockIdx.x * 256 + tid; i < total; i += PC_BLOCKS * 256) {
        int b = i / N_;
        int n = i - b * N_;
        const float* base = pc + (size_t)b * 4 * N_;
        float x = base[n];
        float y = base[N_ + n];
        float z = base[2 * N_ + n];
        const float* M = &sM[b * 12];
        float dx = (M[0] - 1.f) * x + M[1] * y + M[2] * z + M[9];
        float dy = M[3] * x + (M[4] - 1.f) * y + M[5] * z + M[10];
        float dz = M[6] * x + M[7] * y + (M[8] - 1.f) * z + M[11];
        acc += sqrtf(dx*dx + dy*dy + dz*dz);
    }
    acc = waveReduceSum(acc);
    if ((tid & 31) == 0) swave[tid >> 5] = acc;
    __syncthreads();
    if (tid == 0) {
        float s = 0.f;
        #pragma unroll
        for (int w = 0; w < 8; ++w) s += swave[w];
        partials[blockIdx.x] = s;
    }
}

// ---------------------------------------------------------------------------
// Kernel 3: flow loss. One block per iteration t (padded grid: t>=1000 writes
// a zero partial). `valid` tile (shared by both channels) staged into LDS via
// gfx1250 async global->LDS b128 copies; pred/gt streamed with b128 loads.
// Per-block result already weighted by gamma^(T-1-t) and divided by the
// 16384-element mean.
// ---------------------------------------------------------------------------
__global__ void flow_kernel(const float* __restrict__ pred, const float* __restrict__ gt,
                            const float* __restrict__ valid, float* __restrict__ partials) {
    __shared__ __align__(16) float s_valid[B_ * HW_];   // 32 KB
    __shared__ float swave[8];
    const int t = blockIdx.x;
    const int tid = threadIdx.x;
    if (t >= T_) {                       // padding blocks: zero the tail slot
        if (tid == 0) partials[t] = 0.f;
        return;
    }

    // ---- async stage: valid[b, t, 0, :, :] -> LDS, 4 x 8KB tiles -----------
    #pragma unroll
    for (int b = 0; b < B_; ++b) {
        const char* gbase = (const char*)(valid + ((size_t)b * T_ + t) * HW_);
        unsigned lbase = (unsigned)(uintptr_t)(&s_valid[b * HW_]);
        for (int j = tid; j < HW_ / 4; j += 256) {
            unsigned laddr = lbase + (unsigned)j * 16u;
            const char* ga = gbase + (size_t)j * 16;
            asm volatile("global_load_async_to_lds_b128 %0, %1, off"
                         :: "v"(laddr), "v"(ga) : "memory");
        }
    }
    asm volatile("s_wait_asynccnt 0x0" ::: "memory");
    __syncthreads();

    // ---- accumulate sum( valid * |pred - gt| ) over b, c, h, w -------------
    float acc = 0.f;
    #pragma unroll
    for (int b = 0; b < B_; ++b) {
        const float4* p0 = (const float4*)(pred + (((size_t)b * T_ + t) * 2 + 0) * HW_);
        const float4* p1 = (const float4*)(pred + (((size_t)b * T_ + t) * 2 + 1) * HW_);
        const float4* g0 = (const float4*)(gt   + (((size_t)b * T_ + t) * 2 + 0) * HW_);
        const float4* g1 = (const float4*)(gt   + (((size_t)b * T_ + t) * 2 + 1) * HW_);
        const float4* vv = (const float4*)(&s_valid[b * HW_]);
        for (int j = tid; j < HW_ / 4; j += 256) {
            float4 v  = vv[j];
            float4 a0 = p0[j], c0 = g0[j];
            float4 a1 = p1[j], c1 = g1[j];
            acc += v.x * (fabsf(a0.x - c0.x) + fabsf(a1.x - c1.x));
            acc += v.y * (fabsf(a0.y - c0.y) + fabsf(a1.y - c1.y));
            acc += v.z * (fabsf(a0.z - c0.z) + fabsf(a1.z - c1.z));
            acc += v.w * (fabsf(a0.w - c0.w) + fabsf(a1.w - c1.w));
        }
    }
    acc = waveReduceSum(acc);
    if ((tid & 31) == 0) swave[tid >> 5] = acc;
    __syncthreads();
    if (tid == 0) {
        float s = 0.f;
        #pragma unroll
        for (int w = 0; w < 8; ++w) s += swave[w];
        // gamma^(T-1-t) = exp((T-1-t) * ln(0.8))
        float wgt = __expf((float)(T_ - 1 - t) * -0.22314355131420976f);
        partials[t] = wgt * s * (1.0f / (B_ * 2 * HW_));
    }
}

// ---------------------------------------------------------------------------
// Kernel 4: deterministic final combine (single wave). Partial arrays (length
// a multiple of 64) reduced with V_WMMA_F32_16X16X4_F32 against an all-ones B
// matrix: D[m][n]=Σ_k A[m][k]; one wmma folds 64 wave-held values. Branch-free:
// each lane loads its pair as a single b64.
// Total = column 0 = lane0(rows 0-7) + lane16(rows 8-15).
// ---------------------------------------------------------------------------
__device__ __forceinline__ float wmma_sum64(const float* __restrict__ arr, int n64) {
    v2f ones; ones.x = 1.f; ones.y = 1.f;
    v8f acc = {};
    const int lane = threadIdx.x;                 // 0..31, full wave
    const v2f* a2 = (const v2f*)arr;
    for (int c = 0; c < n64; ++c) {
        v2f a = a2[c * 32 + lane];                // global_load_b64, branch-free
        acc = __builtin_amdgcn_wmma_f32_16x16x4_f32(false, a, false, ones,
                                                    (short)0, acc, false, false);
    }
    float s = acc[0] + acc[1] + acc[2] + acc[3] + acc[4] + acc[5] + acc[6] + acc[7];
    return __shfl(s, 0, 32) + __shfl(s, 16, 32);
}

__global__ void final_kernel(const float* __restrict__ ws, float* __restrict__ out) {
    float pc_sum = wmma_sum64(ws + WS_PC,   PC_BLOCKS / 64);  // 8 wmma
    float flow   = wmma_sum64(ws + WS_FLOW, TPAD_ / 64);      // 16 wmma
    if (threadIdx.x == 0) {
        float lt = ws[WS_SCAL + 0];
        float lr = ws[WS_SCAL + 1];
        float pose = lt + lr;                        // RESCALE_TRANS/ROT = 1.0
        float pc_loss = pc_sum / (float)N_;          // Σ_b mean_b
        float pcB = pc_loss / (float)B_;
        float total = 0.5f * pose + 0.5f * pcB + 0.5f * flow;
        out[0] = total;
        out[1] = lt;
        out[2] = lr;
        out[3] = pcB;
        out[4] = flow;
    }
}

// ---------------------------------------------------------------------------
extern "C" void kernel_launch(void* const* d_in, const int* in_sizes, int n_in,
                              void* d_out, int out_size, void* d_ws, size_t ws_size,
                              hipStream_t stream) {
    (void)in_sizes; (void)n_in; (void)out_size; (void)ws_size;
    const float* pc = (const float*)d_in[0];
    const float* tt = (const float*)d_in[1];
    const float* tr = (const float*)d_in[2];
    const float* te = (const float*)d_in[3];
    const float* re = (const float*)d_in[4];
    const float* fp = (const float*)d_in[5];
    const float* fg = (const float*)d_in[6];
    const float* fv = (const float*)d_in[7];
    float* wsf = (float*)d_ws;
    float* out = (float*)d_out;

    setup_kernel<<<1, 32, 0, stream>>>(tt, tr, te, re, wsf);
    pc_kernel<<<PC_BLOCKS, 256, 0, stream>>>(pc, wsf, wsf + WS_PC);
    flow_kernel<<<TPAD_, 256, 0, stream>>>(fp, fg, fv, wsf + WS_FLOW);
    final_kernel<<<1, 32, 0, stream>>>(wsf, out);
}